// UniversalMoEContainer_7602092114452
// MI455X (gfx1250) — compile-verified
//
#include <hip/hip_runtime.h>

// MoE 3x3 conv, top-2 experts, fp32 WMMA (V_WMMA_F32_16X16X4_F32) on gfx1250.
// x halo tile staged with GLOBAL_LOAD_ASYNC_TO_LDS_B32 (ASYNCcnt path).
// B=64, C=CO=64, H=W=56, E=8, K=2.

#define BATCH 64
#define CIN   64
#define COUT  64
#define HH    56
#define WWID  56
#define TOPK  2

typedef float v2f __attribute__((ext_vector_type(2)));
typedef float v8f __attribute__((ext_vector_type(8)));

__global__ __launch_bounds__(256)
void moe_conv3x3_wmma_f32(const float* __restrict__ x,
                          const float* __restrict__ gates,
                          const int*   __restrict__ indices,
                          const float* __restrict__ Wexp,   // [E][CO][C][3][3]
                          const float* __restrict__ bexp,   // [E][CO]
                          float* __restrict__ out)          // [B][CO][H][W]
{
    // 10x10 halo tile for all 64 input channels: 25.6 KB LDS
    __shared__ float xs[CIN * 100];

    const int blk = blockIdx.x;        // B * 49 blocks
    const int b   = blk / 49;
    const int t   = blk % 49;
    const int h0  = (t / 7) * 8;
    const int w0  = (t % 7) * 8;

    const int tid = threadIdx.x;

    // ---- expert selection (uniform per block) + weight prefetch ----
    const int   e0  = indices[b * TOPK + 0];
    const int   e1  = indices[b * TOPK + 1];
    const float gv0 = gates[b * TOPK + 0];
    const float gv1 = gates[b * TOPK + 1];
    __builtin_prefetch(Wexp + (size_t)e0 * COUT * CIN * 9, 0, 1);  // global_prefetch_b8
    __builtin_prefetch(Wexp + (size_t)e1 * COUT * CIN * 9, 0, 1);

    // ---------------- stage padded input tile into LDS (async) ----------------
    // Interior elements: GLOBAL_LOAD_ASYNC_TO_LDS_B32 (memory -> LDS direct,
    // per-lane scatter, tracked by ASYNCcnt). Boundary pad: plain zero store.
    // The two address sets are disjoint, so no LDS write ordering hazard.
    const float* xb = x + (size_t)b * CIN * HH * WWID;
    for (int i = tid; i < CIN * 100; i += 256) {
        int c   = i / 100;
        int rem = i - c * 100;
        int r   = rem / 10;
        int s   = rem - r * 10;
        int gh  = h0 + r - 1;
        int gw  = w0 + s - 1;
        if (gh >= 0 && gh < HH && gw >= 0 && gw < WWID) {
            unsigned int       lds_off = (unsigned int)(unsigned long long)(const void*)(xs + i);
            const float*       gptr    = xb + (c * HH + gh) * WWID + gw;
            asm volatile("global_load_async_to_lds_b32 %0, %1, off"
                         :
                         : "v"(lds_off), "v"(gptr)
                         : "memory");
        } else {
            xs[i] = 0.0f;
        }
    }
#if defined(__has_builtin) && __has_builtin(__builtin_amdgcn_s_wait_asynccnt)
    __builtin_amdgcn_s_wait_asynccnt(0);
#else
    asm volatile("s_wait_asynccnt 0x0" ::: "memory");
#endif
    __syncthreads();

    // ---------------- wave / lane mapping ----------------
    const int lane = tid & 31;
    const int wv   = tid >> 5;
    const int mi   = wv & 3;          // M group: co rows [mi*16, mi*16+16)
    const int nw   = wv >> 2;         // N group: pixels [nw*32, nw*32+32)
    const int l15  = lane & 15;
    const int h    = lane >> 4;       // lane half (selects K pair)

    const int n0   = nw * 32 + l15;           // acc0 pixel index 0..63
    const int n1   = n0 + 16;                 // acc1 pixel index
    const int pix0 = (n0 >> 3) * 10 + (n0 & 7);   // LDS offset of output pixel
    const int pix1 = (n1 >> 3) * 10 + (n1 & 7);

    const int mrow  = mi * 16 + l15;          // A-matrix row (co) for this lane
    const int corow = mi * 16 + 8 * h;        // first C/D row for this lane

    float outv0[8], outv1[8];
    #pragma unroll
    for (int v = 0; v < 8; ++v) { outv0[v] = 0.f; outv1[v] = 0.f; }

    // tap offset table: off(t) = (t/3)*10 + t%3 for t in 0..8
    //   s=0: h=0 -> taps(0,1)=off(0,1)    h=1 -> taps(2,3)=off(2,10)
    //   s=1: h=0 -> taps(4,5)=off(11,12)  h=1 -> taps(6,7)=off(20,21)
    //   s=2: h=0 -> taps(8,pad)=off(22,-) h=1 -> both pad (A=0)
    const int s0o0 = h ? 2  : 0,  s0o1 = h ? 10 : 1;
    const int s1o0 = h ? 20 : 11, s1o1 = h ? 21 : 12;
    const int kofs = 2 * h;                    // K sub-offset within a step

    for (int k = 0; k < TOPK; ++k) {
        const int   e  = k == 0 ? e0 : e1;
        const float gv = k == 0 ? gv0 : gv1;
        const float* Wp = Wexp + ((size_t)e * COUT + mrow) * (CIN * 9);
        const float* bp = bexp + e * COUT + corow;

        // init accumulators with bias (C/D row = corow + v)
        v8f acc0, acc1;
        #pragma unroll
        for (int v = 0; v < 8; ++v) { float bb = bp[v]; acc0[v] = bb; acc1[v] = bb; }

        #pragma unroll 2
        for (int c = 0; c < CIN; ++c) {
            const int cb = c * 100;
            const int kb = c * 9;
            // ---- K step s = 0 (taps 0..3) ----
            {
                v2f a, b0, b1;
                a.x = Wp[kb + kofs];
                a.y = Wp[kb + kofs + 1];
                b0.x = xs[cb + pix0 + s0o0]; b0.y = xs[cb + pix0 + s0o1];
                b1.x = xs[cb + pix1 + s0o0]; b1.y = xs[cb + pix1 + s0o1];
                acc0 = __builtin_amdgcn_wmma_f32_16x16x4_f32(false, a, false, b0, (short)0, acc0, false, false);
                acc1 = __builtin_amdgcn_wmma_f32_16x16x4_f32(false, a, false, b1, (short)0, acc1, false, false);
            }
            // ---- K step s = 1 (taps 4..7) ----
            {
                v2f a, b0, b1;
                a.x = Wp[kb + 4 + kofs];
                a.y = Wp[kb + 5 + kofs];
                b0.x = xs[cb + pix0 + s1o0]; b0.y = xs[cb + pix0 + s1o1];
                b1.x = xs[cb + pix1 + s1o0]; b1.y = xs[cb + pix1 + s1o1];
                acc0 = __builtin_amdgcn_wmma_f32_16x16x4_f32(false, a, false, b0, (short)0, acc0, false, false);
                acc1 = __builtin_amdgcn_wmma_f32_16x16x4_f32(false, a, false, b1, (short)0, acc1, false, false);
            }
            // ---- K step s = 2 (tap 8 + 3 zero-pad taps) ----
            {
                v2f a, b0, b1;
                float a8 = Wp[kb + 8];
                a.x = h ? 0.f : a8;            // h==1 lanes carry only pad taps
                a.y = 0.f;                     // pad
                float bx0 = xs[cb + pix0 + 22];
                float bx1 = xs[cb + pix1 + 22];
                b0.x = bx0; b0.y = bx0;        // pad B values: finite, killed by A=0
                b1.x = bx1; b1.y = bx1;
                acc0 = __builtin_amdgcn_wmma_f32_16x16x4_f32(false, a, false, b0, (short)0, acc0, false, false);
                acc1 = __builtin_amdgcn_wmma_f32_16x16x4_f32(false, a, false, b1, (short)0, acc1, false, false);
            }
        }

        // epilogue: ReLU + gate, accumulate both experts in registers
        #pragma unroll
        for (int v = 0; v < 8; ++v) {
            float y0 = acc0[v]; y0 = y0 > 0.f ? y0 : 0.f;
            float y1 = acc1[v]; y1 = y1 > 0.f ? y1 : 0.f;
            outv0[v] += gv * y0;
            outv1[v] += gv * y1;
        }
    }

    // ---------------- store ----------------
    float* ob = out + (size_t)b * COUT * HH * WWID;
    const int hh0 = h0 + (n0 >> 3), ww0 = w0 + (n0 & 7);
    const int hh1 = h0 + (n1 >> 3), ww1 = w0 + (n1 & 7);
    #pragma unroll
    for (int v = 0; v < 8; ++v) {
        int co = corow + v;
        ob[(co * HH + hh0) * WWID + ww0] = outv0[v];
        ob[(co * HH + hh1) * WWID + ww1] = outv1[v];
    }
}

extern "C" void kernel_launch(void* const* d_in, const int* in_sizes, int n_in,
                              void* d_out, int out_size, void* d_ws, size_t ws_size,
                              hipStream_t stream) {
    const float* x    = (const float*)d_in[0];   // [B,C,H,W]
    const float* wts  = (const float*)d_in[1];   // [B,K] gates
    const int*   idx  = (const int*)  d_in[2];   // [B,K]
    const float* Wexp = (const float*)d_in[3];   // [E,CO,C,3,3]
    const float* bexp = (const float*)d_in[4];   // [E,CO]
    float* out = (float*)d_out;                  // [B,CO,H,W]

    dim3 grid(BATCH * 49);   // 7x7 tiles of 8x8 pixels per sample
    dim3 block(256);
    moe_conv3x3_wmma_f32<<<grid, block, 0, stream>>>(x, wts, idx, Wexp, bexp, out);
}